// LightGCN_28681791602975
// MI455X (gfx1250) — compile-verified
//
#include <hip/hip_runtime.h>
#include <hip/hip_bf16.h>
#include <stdint.h>

// ---------------------------------------------------------------------------
// LightGCN propagation for MI455X (gfx1250, wave32).
//   acc = (E0 + A E0 + A^2 E0 + A^3 E0) / 4,  A given as COO (rows, cols, vals)
//
// Roofline: 1.23 GFLOP total -> trivially memory bound. Embedding table
// (51.2 MB) is resident in the 192 MB L2; edge stream is ~115 MB from HBM
// (~5us at 23.3 TB/s). The binding resource is L2 atomic RMW throughput, so
// we convert COO->CSR on device once (~7M int atomics) and run the 3 SpMM
// layers as pure gather + register-accumulate + one store per row (zero fp32
// atomics in the hot path). CSR payload is packed (col, val) int2 so the hot
// loop issues a single b64 broadcast load per edge. CDNA5 async global->LDS
// DMA (ASYNCcnt) stages the edge tiles for the build/fallback kernels.
// ---------------------------------------------------------------------------

#define N_USERS  100000
#define N_ITEMS  100000
#define N_NODES  (N_USERS + N_ITEMS)
#define EMB_DIM  64
#define NNZ      3200000
#define N_LAYERS 3

#define TPB    256     // 8 wave32s per block
#define EPB    512     // edges staged per block (NNZ % EPB == 0)
#define SCAN_T 1024    // single-block scan width

// Low 32 bits of a generic pointer into LDS == workgroup-relative LDS byte
// address (ISA: flat aperture check truncates to the low 32 address bits).
__device__ __forceinline__ uint32_t lds_addr(const void* p) {
    return (uint32_t)(uintptr_t)p;
}

// CDNA5 async DMA: copy 8 bytes global -> LDS, tracked with ASYNCcnt.
__device__ __forceinline__ void async_copy_b64_to_lds(uint32_t lds_off, const void* gaddr) {
    asm volatile("global_load_async_to_lds_b64 %0, %1, off"
                 :: "v"(lds_off), "v"(gaddr)
                 : "memory");
}

__device__ __forceinline__ void wait_async_lds() {
    asm volatile("s_wait_asynccnt 0x0" ::: "memory");
    __syncthreads();
}

__device__ __forceinline__ void atomic_add_f32(float* p, float v) {
    __hip_atomic_fetch_add(p, v, __ATOMIC_RELAXED, __HIP_MEMORY_SCOPE_AGENT);
}

// ---------------------------------------------------------------------------
// Common: acc = cur = concat(user_emb, item_emb), float4 (global_load_b128).
// ---------------------------------------------------------------------------
__global__ __launch_bounds__(TPB) void lgcn_init_kernel(
    const float4* __restrict__ ue, const float4* __restrict__ ie,
    float4* __restrict__ cur, float4* __restrict__ acc)
{
    const long i   = (long)blockIdx.x * TPB + threadIdx.x;   // float4 index
    const long nu4 = (long)N_USERS * (EMB_DIM / 4);
    const float4 v = (i < nu4) ? ue[i] : ie[i - nu4];
    cur[i] = v;
    acc[i] = v;
}

// ---------------------------------------------------------------------------
// CSR build: zero counts -> histogram -> exclusive scan -> scatter.
// ---------------------------------------------------------------------------
__global__ __launch_bounds__(TPB) void lgcn_zero_cnt_kernel(int* __restrict__ cnt)
{
    const int i = blockIdx.x * TPB + threadIdx.x;
    if (i < N_NODES) cnt[i] = 0;
}

__global__ __launch_bounds__(TPB) void lgcn_hist_kernel(
    const int* __restrict__ rows, int* __restrict__ cnt)
{
    const long e = (long)blockIdx.x * TPB + threadIdx.x;
    atomicAdd(&cnt[rows[e]], 1);
}

// One block of 1024 threads: chunked exclusive scan of cnt[N] -> row_ptr[N+1],
// plus a second copy into fill[] (running write cursors for the scatter).
__global__ __launch_bounds__(SCAN_T) void lgcn_scan_kernel(
    const int* __restrict__ cnt, int* __restrict__ row_ptr, int* __restrict__ fill)
{
    __shared__ int part[SCAN_T];
    const int t  = threadIdx.x;
    const int CH = (N_NODES + SCAN_T - 1) / SCAN_T;   // 196
    const int lo = t * CH;
    const int hi = (lo + CH < N_NODES) ? (lo + CH) : N_NODES;

    int s = 0;
    for (int i = lo; i < hi; ++i) s += cnt[i];
    part[t] = s;
    __syncthreads();

    // Hillis-Steele inclusive scan over the 1024 chunk sums.
    for (int off = 1; off < SCAN_T; off <<= 1) {
        const int v   = part[t];
        const int add = (t >= off) ? part[t - off] : 0;
        __syncthreads();
        part[t] = v + add;
        __syncthreads();
    }

    int run = (t == 0) ? 0 : part[t - 1];             // exclusive chunk prefix
    for (int i = lo; i < hi; ++i) {
        row_ptr[i] = run;
        fill[i]    = run;
        run += cnt[i];
    }
    if (t == SCAN_T - 1) row_ptr[N_NODES] = part[SCAN_T - 1];  // == NNZ
}

// Scatter COO -> packed CSR payload. Edge triples staged via async LDS DMA.
// epak[p] = (col, bitcast(val)) -> one b64 store per edge.
__global__ __launch_bounds__(TPB) void lgcn_scatter_kernel(
    const int* __restrict__ rows, const int* __restrict__ cols,
    const float* __restrict__ vals,
    int* __restrict__ fill, int2* __restrict__ epak)
{
    __shared__ int   s_rows[EPB];
    __shared__ int   s_cols[EPB];
    __shared__ float s_vals[EPB];

    const int  t    = threadIdx.x;
    const long base = (long)blockIdx.x * EPB;
    const int  e2   = 2 * t;
    async_copy_b64_to_lds(lds_addr(&s_rows[e2]), rows + base + e2);
    async_copy_b64_to_lds(lds_addr(&s_cols[e2]), cols + base + e2);
    async_copy_b64_to_lds(lds_addr(&s_vals[e2]), vals + base + e2);
    wait_async_lds();

    #pragma unroll
    for (int k = 0; k < EPB / TPB; ++k) {
        const int e = t + k * TPB;
        const int r = s_rows[e];
        const int p = atomicAdd(&fill[r], 1);
        epak[p] = make_int2(s_cols[e], __float_as_int(s_vals[e]));
    }
}

// ---------------------------------------------------------------------------
// CSR SpMM, fused accumulate: per row r (16 lanes x float4):
//   nxt[r] = sum_j val_j * cur[col_j];  acc[r] = (acc[r] + nxt[r]) * scale
// Zero fp32 atomics: one b64 edge load + one b128 gather per edge, register
// accumulation, one b128 store per lane.
// ---------------------------------------------------------------------------
__global__ __launch_bounds__(TPB) void lgcn_spmm_csr_kernel(
    const int* __restrict__ row_ptr, const int2* __restrict__ epak,
    const float* __restrict__ cur,
    float* __restrict__ nxt, float* __restrict__ acc, float scale)
{
    const int t  = threadIdx.x;
    const int g  = t >> 4;                       // 16 rows per block
    const int l  = t & 15;                       // 16 lanes per row
    const int r  = blockIdx.x * 16 + g;          // grid = N_NODES/16 exactly
    const int c4 = l * 4;

    const int start = row_ptr[r];
    const int end   = row_ptr[r + 1];

    float4 s = make_float4(0.f, 0.f, 0.f, 0.f);
    for (int j = start; j < end; ++j) {
        const int2  e = epak[j];                 // b64 broadcast across 16 lanes
        const int   c = e.x;
        const float v = __int_as_float(e.y);
        const float4 x = *reinterpret_cast<const float4*>(
                             cur + (size_t)c * EMB_DIM + c4);   // L2-resident
        s.x += v * x.x; s.y += v * x.y; s.z += v * x.z; s.w += v * x.w;
    }

    const size_t o = (size_t)r * EMB_DIM + c4;
    *reinterpret_cast<float4*>(nxt + o) = s;

    float4 a = *reinterpret_cast<const float4*>(acc + o);
    a.x = (a.x + s.x) * scale;
    a.y = (a.y + s.y) * scale;
    a.z = (a.z + s.z) * scale;
    a.w = (a.w + s.w) * scale;
    *reinterpret_cast<float4*>(acc + o) = a;
}

// ---------------------------------------------------------------------------
// Fallback path (small workspace): COO SpMM with device-scope f32 atomics.
// ---------------------------------------------------------------------------
__global__ __launch_bounds__(TPB) void lgcn_zero_kernel(float4* __restrict__ p)
{
    const long i = (long)blockIdx.x * TPB + threadIdx.x;
    p[i] = make_float4(0.f, 0.f, 0.f, 0.f);
}

__global__ __launch_bounds__(TPB) void lgcn_spmm_atomic_kernel(
    const int*   __restrict__ rows, const int* __restrict__ cols,
    const float* __restrict__ vals, const float* __restrict__ cur,
    float*       __restrict__ nxt)
{
    __shared__ int   s_rows[EPB];
    __shared__ int   s_cols[EPB];
    __shared__ float s_vals[EPB];

    const int  t    = threadIdx.x;
    const long base = (long)blockIdx.x * EPB;
    const int  e2   = 2 * t;
    async_copy_b64_to_lds(lds_addr(&s_rows[e2]), rows + base + e2);
    async_copy_b64_to_lds(lds_addr(&s_cols[e2]), cols + base + e2);
    async_copy_b64_to_lds(lds_addr(&s_vals[e2]), vals + base + e2);
    wait_async_lds();

    const int g  = t >> 4;
    const int l  = t & 15;
    const int c4 = l * 4;

    #pragma unroll 4
    for (int k = 0; k < EPB / 16; ++k) {
        const int   e = g * (EPB / 16) + k;
        const int   r = s_rows[e];
        const int   c = s_cols[e];
        const float v = s_vals[e];
        const float4 x = *reinterpret_cast<const float4*>(
                             cur + (size_t)c * EMB_DIM + c4);
        float* dst = nxt + (size_t)r * EMB_DIM + c4;
        atomic_add_f32(dst + 0, v * x.x);
        atomic_add_f32(dst + 1, v * x.y);
        atomic_add_f32(dst + 2, v * x.z);
        atomic_add_f32(dst + 3, v * x.w);
    }
}

__global__ __launch_bounds__(TPB) void lgcn_accum_kernel(
    float4* __restrict__ acc, const float4* __restrict__ nxt, float scale)
{
    const long i = (long)blockIdx.x * TPB + threadIdx.x;
    float4 a = acc[i];
    const float4 b = nxt[i];
    a.x = (a.x + b.x) * scale;
    a.y = (a.y + b.y) * scale;
    a.z = (a.z + b.z) * scale;
    a.w = (a.w + b.w) * scale;
    acc[i] = a;
}

// ---------------------------------------------------------------------------
extern "C" void kernel_launch(void* const* d_in, const int* in_sizes, int n_in,
                              void* d_out, int out_size, void* d_ws, size_t ws_size,
                              hipStream_t stream) {
    const float* ue   = (const float*)d_in[0];
    const float* ie   = (const float*)d_in[1];
    const int*   rows = (const int*)  d_in[2];
    const int*   cols = (const int*)  d_in[3];
    const float* vals = (const float*)d_in[4];

    float* acc = (float*)d_out;                       // 200000 x 64 fp32

    const size_t S = (size_t)N_NODES * EMB_DIM * sizeof(float);  // 51.2 MB
    char*  w   = (char*)d_ws;
    float* cur = (float*)w;                           // ping
    float* nxt = (float*)(w + S);                     // pong

    const int  n4 = N_NODES * (EMB_DIM / 4);          // 3.2M float4 elements
    const dim3 gElem(n4 / TPB);                       // 12500
    const dim3 gEdgeT(NNZ / TPB);                     // 12500 (thread-per-edge)
    const dim3 gEdgeB(NNZ / EPB);                     // 6250  (block tiles)
    const dim3 gRows(N_NODES / 16);                   // 12500 (16 rows/block)
    const dim3 gNode((N_NODES + TPB - 1) / TPB);      // 782

    // CSR workspace layout after the two ping-pong tables (epak, row_ptr 8B-aligned; fill, cnt 4B-aligned).
    int2*  epak    = (int2*)(w + 2 * S);                          // NNZ * 8B
    int*   row_ptr = (int*) (w + 2 * S + (size_t)NNZ * 8);        // (N+1) * 4B
    int*   fill    = row_ptr + (N_NODES + 1);                     // N * 4B
    int*   cnt     = fill + N_NODES;                              // N * 4B
    const size_t NEED_CSR = 2 * S + (size_t)NNZ * 8
                          + (size_t)(3 * N_NODES + 1) * 4;        // ~130 MB

    lgcn_init_kernel<<<gElem, TPB, 0, stream>>>(
        (const float4*)ue, (const float4*)ie, (float4*)cur, (float4*)acc);

    if (ws_size >= NEED_CSR) {
        // ---- Fast path: build CSR once, then atomic-free gather SpMM. ----
        lgcn_zero_cnt_kernel<<<gNode, TPB, 0, stream>>>(cnt);
        lgcn_hist_kernel<<<gEdgeT, TPB, 0, stream>>>(rows, cnt);
        lgcn_scan_kernel<<<1, SCAN_T, 0, stream>>>(cnt, row_ptr, fill);
        lgcn_scatter_kernel<<<gEdgeB, TPB, 0, stream>>>(
            rows, cols, vals, fill, epak);

        for (int layer = 0; layer < N_LAYERS; ++layer) {
            const float scale = (layer == N_LAYERS - 1) ? 0.25f : 1.0f;
            lgcn_spmm_csr_kernel<<<gRows, TPB, 0, stream>>>(
                row_ptr, epak, cur, nxt, acc, scale);
            float* tmp = cur; cur = nxt; nxt = tmp;   // host swap: deterministic
        }
    } else {
        // ---- Fallback: COO SpMM with device-scope fp32 atomics. ----
        for (int layer = 0; layer < N_LAYERS; ++layer) {
            lgcn_zero_kernel<<<gElem, TPB, 0, stream>>>((float4*)nxt);
            lgcn_spmm_atomic_kernel<<<gEdgeB, TPB, 0, stream>>>(
                rows, cols, vals, cur, nxt);
            const float scale = (layer == N_LAYERS - 1) ? 0.25f : 1.0f;
            lgcn_accum_kernel<<<gElem, TPB, 0, stream>>>(
                (float4*)acc, (const float4*)nxt, scale);
            float* tmp = cur; cur = nxt; nxt = tmp;
        }
    }
}